// GetDensity_25512105739132
// MI455X (gfx1250) — compile-verified
//
#include <hip/hip_runtime.h>
#include <cmath>

// ---------------- problem constants (from reference) ----------------
#define NTYPE 4
#define NWAVE 8
#define NIPS  3
#define NCH   13          // 1 + 3 + 9 angular channels
#define DDIM  (NWAVE*NIPS) // 24
#define SWROW (NCH*NWAVE)  // 104

typedef __attribute__((ext_vector_type(2))) float v2f;
typedef __attribute__((ext_vector_type(8))) float v8f;

// ---------------- helper kernels ----------------
__global__ void zero_kernel(float* __restrict__ p, int n) {
    int i = blockIdx.x * blockDim.x + threadIdx.x;
    if (i < n) p[i] = 0.0f;
}

__global__ void init_orb_kernel(const float* __restrict__ params, // (NTYPE,24)
                                const int* __restrict__ species,
                                float* __restrict__ orb, int A) {
    int i = blockIdx.x * blockDim.x + threadIdx.x;
    if (i >= A * DDIM) return;
    int a = i / DDIM, d = i - a * DDIM;
    orb[i] = params[species[a] * DDIM + d];
}

// ---------------- per-pair accumulate: w -> atomic sw ----------------
// w[p,j,m] = ang[j] * cut * (radial . hyp[b])[m] * orb[jn, b*8+m],  b = index_para[j]
__global__ void pair_accum_kernel(const float* __restrict__ cart,
                                  const float* __restrict__ shifts,
                                  const float* __restrict__ rs,     // (NTYPE,8)
                                  const float* __restrict__ inta,   // (NTYPE,8)
                                  const int*   __restrict__ neigh,  // (2,P)
                                  const int*   __restrict__ species,
                                  const float* __restrict__ hyp,    // (3,8,8) slice
                                  const float* __restrict__ orb,    // (A,24)
                                  float* __restrict__ sw,           // (A,13,8)
                                  int P) {
    int p = blockIdx.x * blockDim.x + threadIdx.x;
    if (p >= P) return;
    int ic = neigh[p];
    int jn = neigh[P + p];
    float dx = cart[3*ic+0] - cart[3*jn+0] - shifts[3*p+0];
    float dy = cart[3*ic+1] - cart[3*jn+1] - shifts[3*p+1];
    float dz = cart[3*ic+2] - cart[3*jn+2] - shifts[3*p+2];
    float dist = sqrtf(dx*dx + dy*dy + dz*dz);
    float inv  = 1.0f / dist;
    float ux = dx*inv, uy = dy*inv, uz = dz*inv;
    float c  = 0.5f * cosf(dist * 0.6283185307179586f) + 0.5f; // pi/5
    float cut = c * c;
    int sp = species[jn];

    float radial[NWAVE];
#pragma unroll
    for (int k = 0; k < NWAVE; ++k) {
        float t = dist - rs[sp*NWAVE + k];
        radial[k] = expf(-inta[sp*NWAVE + k] * t * t);
    }

    float T[NIPS][NWAVE];
#pragma unroll
    for (int b = 0; b < NIPS; ++b) {
#pragma unroll
        for (int m = 0; m < NWAVE; ++m) {
            float r = 0.0f;
#pragma unroll
            for (int k = 0; k < NWAVE; ++k)
                r = fmaf(radial[k], hyp[b*64 + k*8 + m], r);
            T[b][m] = cut * r * orb[jn*DDIM + b*NWAVE + m];
        }
    }

    float ang[NCH] = {1.0f, ux, uy, uz,
                      ux*ux, ux*uy, ux*uz,
                      uy*ux, uy*uy, uy*uz,
                      uz*ux, uz*uy, uz*uz};

    float* dst = sw + (size_t)ic * SWROW;
#pragma unroll
    for (int j = 0; j < NCH; ++j) {
        const int b = (j == 0) ? 0 : (j < 4 ? 1 : 2);
#pragma unroll
        for (int m = 0; m < NWAVE; ++m)
            unsafeAtomicAdd(&dst[j*NWAVE + m], ang[j] * T[b][m]);  // global_atomic_add_f32, L2-resident
    }
}

// ---------------- density: square + reduce channels by block ----------------
__global__ void density_kernel(const float* __restrict__ sw, // (A,13,8)
                               float* __restrict__ density,  // (A,24)
                               int A) {
    int i = blockIdx.x * blockDim.x + threadIdx.x;
    if (i >= A * NWAVE) return;
    int a = i >> 3, m = i & 7;
    const float* row = sw + (size_t)a * SWROW;
    float d0 = row[m] * row[m];
    float d1 = 0.0f, d2 = 0.0f;
#pragma unroll
    for (int j = 1; j < 4; ++j)  { float v = row[j*NWAVE + m]; d1 = fmaf(v, v, d1); }
#pragma unroll
    for (int j = 4; j < 13; ++j) { float v = row[j*NWAVE + m]; d2 = fmaf(v, v, d2); }
    density[a*DDIM +           m] = d0;
    density[a*DDIM +   NWAVE + m] = d1;
    density[a*DDIM + 2*NWAVE + m] = d2;
}

// ---------------- FP32 WMMA update: orb_out = tanh(density@W + emb[sp]) + orb_in ----------------
// One wave per 16-atom tile; V_WMMA_F32_16X16X4_F32, 2 N-tiles x 6 K-steps.
__global__ void update_wmma_kernel(const float* __restrict__ density, // (A,24)
                                   const float* __restrict__ W,       // (24,24)
                                   const float* __restrict__ emb,     // (NTYPE,24)
                                   const int*   __restrict__ species,
                                   const float* __restrict__ orb_in,  // (A,24)
                                   float* __restrict__ orb_out,       // (A,24)
                                   int A) {
    int wave = (blockIdx.x * blockDim.x + threadIdx.x) >> 5;
    int lane = threadIdx.x & 31;
    int base = wave * 16;
    if (base >= A) return;                 // wave-uniform: EXEC stays all-1s for WMMA

    const int row = lane & 15;             // M row (A frag) / N col (B frag)
    const int hi  = lane >> 4;             // selects K pair within 4-wide K step
    const int atom = base + row;
    const bool rowValid = atom < A;

    // A fragments: lane<16 holds A[row][kb+0..1], lane>=16 holds A[row][kb+2..3]
    v2f afrag[6];
#pragma unroll
    for (int s = 0; s < 6; ++s) {
        int kb = 4*s + 2*hi;
        afrag[s].x = rowValid ? density[atom*DDIM + kb    ] : 0.0f;
        afrag[s].y = rowValid ? density[atom*DDIM + kb + 1] : 0.0f;
    }

#pragma unroll
    for (int t = 0; t < 2; ++t) {
        const int nbase = t * 16;
        const int ncol  = nbase + row;
        const bool colValid = ncol < DDIM;

        v8f acc = {};
#pragma unroll
        for (int s = 0; s < 6; ++s) {
            int kb = 4*s + 2*hi;
            v2f b;
            b.x = colValid ? W[(kb    ) * DDIM + ncol] : 0.0f;
            b.y = colValid ? W[(kb + 1) * DDIM + ncol] : 0.0f;
            acc = __builtin_amdgcn_wmma_f32_16x16x4_f32(
                      false, afrag[s], false, b, (short)0, acc, false, false);
        }
        // D layout: VGPR r -> lanes0-15: D[r][lane], lanes16-31: D[r+8][lane-16]
#pragma unroll
        for (int r = 0; r < 8; ++r) {
            int m  = r + 8*hi;
            int a2 = base + m;
            int n  = nbase + row;
            if (a2 < A && n < DDIM) {
                int sp = species[a2];
                float v = tanhf(acc[r] + emb[sp*DDIM + n]) + orb_in[a2*DDIM + n];
                orb_out[a2*DDIM + n] = v;
            }
        }
    }
}

// ---------------- launch ----------------
extern "C" void kernel_launch(void* const* d_in, const int* in_sizes, int n_in,
                              void* d_out, int out_size, void* d_ws, size_t ws_size,
                              hipStream_t stream) {
    const float* cart    = (const float*)d_in[0];
    const float* shifts  = (const float*)d_in[1];
    const float* rs      = (const float*)d_in[2];
    const float* inta    = (const float*)d_in[3];
    const float* params  = (const float*)d_in[4];
    const float* hyper   = (const float*)d_in[5]; // (3,3,8,8), pre-normalized
    const float* ocW     = (const float*)d_in[6]; // (2,24,24)
    const float* ocEmb   = (const float*)d_in[7]; // (2,4,24)
    const int*   neigh   = (const int*)d_in[8];   // (2,P)
    const int*   species = (const int*)d_in[9];   // (A,)

    const int A = in_sizes[9];
    const int P = in_sizes[1] / 3;

    float* ws   = (float*)d_ws;
    float* sw   = ws;                              // A*104
    float* orb0 = sw   + (size_t)A * SWROW;        // A*24
    float* orb1 = orb0 + (size_t)A * DDIM;
    float* orb2 = orb1 + (size_t)A * DDIM;
    float* den0 = orb2 + (size_t)A * DDIM;
    float* den1 = den0 + (size_t)A * DDIM;

    const int TB = 256;
    dim3 blk(TB);
    dim3 gP((P + TB - 1) / TB);
    dim3 gSW((A * SWROW + TB - 1) / TB);
    dim3 gOrb((A * DDIM + TB - 1) / TB);
    dim3 gDen((A * NWAVE + TB - 1) / TB);
    int nWaves = (A + 15) / 16;
    dim3 gUpd((nWaves * 32 + 127) / 128);
    dim3 blkUpd(128);

    float* orbs[3] = {orb0, orb1, orb2};
    float* dens[3] = {den0, den1, (float*)d_out};

    init_orb_kernel<<<gOrb, blk, 0, stream>>>(params, species, orb0, A);

    for (int it = 0; it < 3; ++it) {
        zero_kernel<<<gSW, blk, 0, stream>>>(sw, A * SWROW);
        pair_accum_kernel<<<gP, blk, 0, stream>>>(
            cart, shifts, rs, inta, neigh, species,
            hyper + (size_t)it * NIPS * NWAVE * NWAVE,
            orbs[it], sw, P);
        density_kernel<<<gDen, blk, 0, stream>>>(sw, dens[it], A);
        if (it < 2) {
            update_wmma_kernel<<<gUpd, blkUpd, 0, stream>>>(
                dens[it], ocW + (size_t)it * DDIM * DDIM,
                ocEmb + (size_t)it * NTYPE * DDIM,
                species, orbs[it], orbs[it + 1], A);
        }
    }
}